// DL_GNN_24979529793811
// MI455X (gfx1250) — compile-verified
//
#include <hip/hip_runtime.h>
#include <hip/hip_bf16.h>

typedef float v2f __attribute__((ext_vector_type(2)));
typedef float v8f __attribute__((ext_vector_type(8)));

#define GCN_IN_CH  256
#define GCN_HIDDEN 64
#define GCN_OUT_CH 128

// ---------------------------------------------------------------------------
// GEMM: C[M, NT*16] = A[M,K] @ B[K, NT*16], f32 row-major,
// via V_WMMA_F32_16X16X4_F32. One wave per 16-row M strip, NT accumulators.
// A fragment loaded once per k-step (non-temporal: streamed exactly once),
// then reused across NT WMMAs. B is tiny (<=128KB) and stays cache-hot.
// Frag layouts (ISA 7.12.2):
//   A 16x4 : lanes 0-15 -> K=k0,k0+1 ; lanes 16-31 -> K=k0+2,k0+3 (float2 load)
//   B 4x16 : mirrored; C/D: VGPR i -> M = i + (lane>=16 ? 8 : 0), N = lane&15
// ---------------------------------------------------------------------------
template <int NT>
__global__ void gcn_gemm_wmma_f32(const float* __restrict__ A,
                                  const float* __restrict__ B,
                                  float* __restrict__ C,
                                  int M, int K) {
    constexpr int N = NT * 16;
    const int lane = threadIdx.x & 31;
    const int tm   = (int)((blockIdx.x * blockDim.x + threadIdx.x) >> 5); // wave = M tile
    if (tm * 16 >= M) return;                 // wave-uniform: EXEC all-1s for WMMA

    const int mr   = tm * 16 + (lane & 15);
    const int nc0  = lane & 15;
    const int koff = (lane >> 4) << 1;        // 0 or 2

    const float* __restrict__ arow = A + (size_t)mr * K + koff;
    const float* __restrict__ bp   = B + (size_t)koff * N + nc0;

    v8f acc[NT];
    #pragma unroll
    for (int t = 0; t < NT; ++t) acc[t] = (v8f){};

    #pragma unroll 4
    for (int k0 = 0; k0 < K; k0 += 4) {
        v2f a = __builtin_nontemporal_load((const v2f*)(arow + k0)); // streamed once
        const float* __restrict__ b0 = bp + (size_t)k0 * N;
        #pragma unroll
        for (int t = 0; t < NT; ++t) {
            v2f b;
            b.x = b0[t * 16];
            b.y = b0[t * 16 + N];
            acc[t] = __builtin_amdgcn_wmma_f32_16x16x4_f32(
                false, a, false, b, (short)0, acc[t], false, false);
        }
    }

    const int row0 = tm * 16 + ((lane >> 4) << 3);
    #pragma unroll
    for (int t = 0; t < NT; ++t) {
        #pragma unroll
        for (int i = 0; i < 8; ++i)
            C[(size_t)(row0 + i) * N + t * 16 + nc0] = acc[t][i];
    }
}

// deg[col[e]] += 1  (f32 counts; self-loop +1 folded into dinv kernel)
__global__ void gcn_degree(const int* __restrict__ col, float* __restrict__ deg, int E) {
    int e = blockIdx.x * blockDim.x + threadIdx.x;
    if (e < E) atomicAdd(&deg[col[e]], 1.0f);
}

// dinv[i] = rsqrt(in_deg(i) + 1)   (in place over the deg buffer)
__global__ void gcn_dinv(float* __restrict__ deg, int Nn) {
    int i = blockIdx.x * blockDim.x + threadIdx.x;
    if (i < Nn) deg[i] = rsqrtf(deg[i] + 1.0f);
}

// One wave per edge: indices/norm fetched once (coalesced), lanes stride channels.
// out[col[e], c] += h[row[e], c] * dinv[row]*dinv[col]
template <int F>
__global__ void gcn_aggregate(const float* __restrict__ h,
                              const int* __restrict__ row,
                              const int* __restrict__ col,
                              const float* __restrict__ dinv,
                              float* __restrict__ out, int E) {
    const int lane = threadIdx.x & 31;
    const int e    = (int)((blockIdx.x * blockDim.x + threadIdx.x) >> 5);
    if (e >= E) return;
    const int r = row[e];
    const int d = col[e];
    const float norm = dinv[r] * dinv[d];
    const float* __restrict__ hs = h + (size_t)r * F;
    float* __restrict__ od = out + (size_t)d * F;
    #pragma unroll
    for (int c = lane; c < F; c += 32)
        atomicAdd(&od[c], hs[c] * norm);
}

// agg[i,c] = relu(agg[i,c] + h[i,c]*dinv[i]^2 + bias[c])  -- float4 vectorized
template <int F>
__global__ void gcn_bias_relu_self(float* __restrict__ agg,
                                   const float* __restrict__ h,
                                   const float* __restrict__ dinv,
                                   const float* __restrict__ bias, int Nn) {
    constexpr int FV = F / 4;
    long long t = (long long)blockIdx.x * blockDim.x + threadIdx.x;
    int i  = (int)(t / FV);
    int c4 = (int)(t - (long long)i * FV);
    if (i >= Nn) return;
    float di = dinv[i];
    float sl = di * di;
    size_t idx = (size_t)i * FV + c4;
    float4 a = ((const float4*)agg)[idx];
    float4 hv = ((const float4*)h)[idx];
    float4 bv = ((const float4*)bias)[c4];
    float4 o;
    o.x = fmaxf(a.x + hv.x * sl + bv.x, 0.0f);
    o.y = fmaxf(a.y + hv.y * sl + bv.y, 0.0f);
    o.z = fmaxf(a.z + hv.z * sl + bv.z, 0.0f);
    o.w = fmaxf(a.w + hv.w * sl + bv.w, 0.0f);
    ((float4*)agg)[idx] = o;
}

// gsum[c] += partial column sums of h2 (128 channels)
__global__ void gcn_mean_reduce(const float* __restrict__ h2,
                                float* __restrict__ gsum, int Nn) {
    int c  = threadIdx.x & 127;
    int rb = threadIdx.x >> 7;                // 256 threads = 2 node rows
    float s = 0.0f;
    for (int i = blockIdx.x * 2 + rb; i < Nn; i += gridDim.x * 2)
        s += h2[(size_t)i * 128 + c];
    atomicAdd(&gsum[c], s);
}

// out[o] = (1/N) * sum_c gsum[c] * fcW[o,c] + fcb[o],  o in {0,1}
__global__ void gcn_final_fc(const float* __restrict__ gsum,
                             const float* __restrict__ fcW,
                             const float* __restrict__ fcb,
                             float* __restrict__ out, float invN) {
    int o = threadIdx.x;
    if (o < 2) {
        float s = 0.0f;
        for (int c = 0; c < 128; ++c)
            s += gsum[c] * invN * fcW[o * 128 + c];
        out[o] = s + fcb[o];
    }
}

static inline size_t align256(size_t v) { return (v + 255) & ~(size_t)255; }

extern "C" void kernel_launch(void* const* d_in, const int* in_sizes, int n_in,
                              void* d_out, int out_size, void* d_ws, size_t ws_size,
                              hipStream_t stream) {
    (void)n_in; (void)out_size; (void)ws_size;

    const float* x   = (const float*)d_in[0];
    const int*   ei  = (const int*)  d_in[1];
    const float* W1  = (const float*)d_in[2];
    const float* b1  = (const float*)d_in[3];
    const float* W2  = (const float*)d_in[4];
    const float* b2  = (const float*)d_in[5];
    const float* fcW = (const float*)d_in[6];
    const float* fcb = (const float*)d_in[7];

    const int Nn = in_sizes[0] / GCN_IN_CH;   // 100000
    const int E  = in_sizes[1] / 2;           // 3200000
    const int* row = ei;
    const int* col = ei + E;

    // Workspace carve-up (h/agg buffers are L2-resident working sets)
    char* ws = (char*)d_ws;
    size_t off = 0;
    float* dinv = (float*)(ws + off); off += align256((size_t)Nn * 4);
    float* h1   = (float*)(ws + off); off += align256((size_t)Nn * GCN_HIDDEN * 4);
    float* agg1 = (float*)(ws + off); off += align256((size_t)Nn * GCN_HIDDEN * 4);
    float* h2   = (float*)(ws + off); off += align256((size_t)Nn * GCN_OUT_CH * 4);
    float* agg2 = (float*)(ws + off); off += align256((size_t)Nn * GCN_OUT_CH * 4);
    float* gsum = (float*)(ws + off); off += align256(128 * 4);

    hipMemsetAsync(dinv, 0, (size_t)Nn * 4, stream);
    hipMemsetAsync(agg1, 0, (size_t)Nn * GCN_HIDDEN * 4, stream);
    hipMemsetAsync(agg2, 0, (size_t)Nn * GCN_OUT_CH * 4, stream);
    hipMemsetAsync(gsum, 0, 128 * 4, stream);

    // Degrees -> dinv
    gcn_degree<<<(E + 255) / 256, 256, 0, stream>>>(col, dinv, E);
    gcn_dinv<<<(Nn + 255) / 256, 256, 0, stream>>>(dinv, Nn);

    const long long mWaves  = Nn / 16;                      // 6250 strip-waves
    const int       mBlocks = (int)((mWaves * 32 + 255) / 256);

    // Layer 1: h1 = x @ W1  (M=100000, N=64 -> NT=4, K=256)
    gcn_gemm_wmma_f32<GCN_HIDDEN / 16><<<mBlocks, 256, 0, stream>>>(x, W1, h1, Nn, GCN_IN_CH);

    // Edge aggregation (1 wave / edge) + fused self-loop/bias/relu
    {
        long long thr = (long long)E * 32;
        gcn_aggregate<GCN_HIDDEN><<<(int)((thr + 255) / 256), 256, 0, stream>>>(h1, row, col, dinv, agg1, E);
        long long thr2 = (long long)Nn * (GCN_HIDDEN / 4);
        gcn_bias_relu_self<GCN_HIDDEN><<<(int)((thr2 + 255) / 256), 256, 0, stream>>>(agg1, h1, dinv, b1, Nn);
    }

    // Layer 2: h2 = agg1 @ W2  (M=100000, N=128 -> NT=8, K=64)
    gcn_gemm_wmma_f32<GCN_OUT_CH / 16><<<mBlocks, 256, 0, stream>>>(agg1, W2, h2, Nn, GCN_HIDDEN);

    {
        long long thr = (long long)E * 32;
        gcn_aggregate<GCN_OUT_CH><<<(int)((thr + 255) / 256), 256, 0, stream>>>(h2, row, col, dinv, agg2, E);
        long long thr2 = (long long)Nn * (GCN_OUT_CH / 4);
        gcn_bias_relu_self<GCN_OUT_CH><<<(int)((thr2 + 255) / 256), 256, 0, stream>>>(agg2, h2, dinv, b2, Nn);
    }

    // Mean over nodes, then 2-way FC
    gcn_mean_reduce<<<512, 256, 0, stream>>>(agg2, gsum, Nn);
    gcn_final_fc<<<1, 32, 0, stream>>>(gsum, fcW, fcb, (float*)d_out, 1.0f / (float)Nn);
}